// DAGProp_75445395521543
// MI455X (gfx1250) — compile-verified
//
#include <hip/hip_runtime.h>
#include <hip/hip_bf16.h>
#include <math.h>

typedef __attribute__((ext_vector_type(2))) float v2f;
typedef __attribute__((ext_vector_type(4))) float v4f;
typedef __attribute__((ext_vector_type(8))) float v8f;

#define NUM_LEVELS 8
#define NODES_PER_LEVEL 16384
#define CHILDREN 8
#define DIM 64
#define EDGES_PER_LEVEL (NODES_PER_LEVEL * CHILDREN)   // 131072

// -------------------------------------------------------------------------
// Kernel 1: out[n][:] = x[n][:] @ W_r^T   for all N rows.
// One wave computes a 16-row slab (16 x 64). A is cached in VGPRs (16 x v2f
// per lane) and reused for all four 16-col N-tiles. fp32 WMMA 16x16x4.
//
// A layout (32-bit A 16x4): lane = M (mod 16); lane>=16 selects K={2,3},
// regs are consecutive K -> per k-step a lane loads x[(row0+m)*64 + 4*ks + 2*hi].
// B layout mirrors it with N across lanes: B[k][n] = Wr[n][k] -> lane loads
// Wr[(n0+m)*64 + 4*ks + 2*hi].
// C/D layout: reg v, lane -> D[v + 8*hi][n0 + m].
// -------------------------------------------------------------------------
__global__ void __launch_bounds__(256)
dag_root_gemm(const float* __restrict__ x, const float* __restrict__ Wr,
              float* __restrict__ out) {
    const int t    = threadIdx.x;
    const int wave = t >> 5;
    const int lane = t & 31;
    const int m    = lane & 15;
    const int hi   = lane >> 4;                  // 0: K={0,1}, 1: K={2,3}

    const int rowTile = blockIdx.x * 8 + wave;   // 8192 row tiles
    const int row0    = rowTile * 16;

    const float* xrow = x + (size_t)(row0 + m) * DIM;

    v2f a[16];
#pragma unroll
    for (int ks = 0; ks < 16; ++ks)
        a[ks] = *(const v2f*)(xrow + ks * 4 + hi * 2);

#pragma unroll
    for (int nt = 0; nt < 4; ++nt) {
        const int n0 = nt * 16;
        const float* wrow = Wr + (size_t)(n0 + m) * DIM;
        v8f acc = {};
#pragma unroll
        for (int ks = 0; ks < 16; ++ks) {
            v2f b = *(const v2f*)(wrow + ks * 4 + hi * 2);
            acc = __builtin_amdgcn_wmma_f32_16x16x4_f32(
                false, a[ks], false, b, (short)0, acc, false, false);
        }
#pragma unroll
        for (int v = 0; v < 8; ++v) {
            const int row = row0 + v + 8 * hi;
            out[(size_t)row * DIM + n0 + m] = acc[v];
        }
    }
}

// -------------------------------------------------------------------------
// Kernel 2: one level of the DAG sweep (launched for lvl = 1..7).
//   agg[p] = sum of out[child] over the parent's 8 children (CSR layout of src)
//   out[p] = tanh( out[p] + agg[p] @ W_l^T + b_l )     (out[p] holds x@Wr^T)
//
// Block = 256 threads = 32 parents.
//   Phase 1: 8 threads per parent aggregate 64 dims into LDS (stride 66 pad
//            -> conflict-free ds_load_b64 in phase 2).
//   Phase 2: 8 waves = 2 parent-groups x 4 N-tiles of fp32 WMMA.
// -------------------------------------------------------------------------
__global__ void __launch_bounds__(256)
dag_level(float* __restrict__ out, const int* __restrict__ src,
          const float* __restrict__ Wl, const float* __restrict__ bl,
          int lvl) {
    __shared__ float agg[32 * 66];               // 8448 B, padded rows

    const int t  = threadIdx.x;
    const int p0 = blockIdx.x * 32;              // parent base (relative)
    const int ebase = (lvl - 1) * EDGES_PER_LEVEL;

    // ---- Phase 1: aggregate children ----
    {
        const int p     = t >> 3;                // 0..31 parent within block
        const int chunk = t & 7;                 // 16-float dim chunk? no: 8-float
        const int d0    = chunk * 8;             // dims d0 .. d0+7
        const int* cidx = src + ebase + (size_t)(p0 + p) * CHILDREN;
        v4f s0 = {}, s1 = {};
#pragma unroll
        for (int c = 0; c < CHILDREN; ++c) {
            const int child = cidx[c];
            const float* crow = out + (size_t)child * DIM + d0;
            s0 += *(const v4f*)(crow);
            s1 += *(const v4f*)(crow + 4);
        }
        float* dst = &agg[p * 66 + d0];          // 8-byte aligned (66 even)
        *(v2f*)(dst + 0) = v2f{s0.x, s0.y};
        *(v2f*)(dst + 2) = v2f{s0.z, s0.w};
        *(v2f*)(dst + 4) = v2f{s1.x, s1.y};
        *(v2f*)(dst + 6) = v2f{s1.z, s1.w};
    }
    __syncthreads();

    // ---- Phase 2: WMMA tiles ----
    const int wave = t >> 5;
    const int lane = t & 31;
    const int pg   = wave >> 2;                  // parent group 0/1 (16 parents)
    const int nt   = wave & 3;                   // N tile
    const int n0   = nt * 16;
    const int m    = lane & 15;
    const int hi   = lane >> 4;

    const float* arow = &agg[(pg * 16 + m) * 66];
    const float* wrow = Wl + (size_t)(n0 + m) * DIM;

    v8f acc = {};
#pragma unroll
    for (int ks = 0; ks < 16; ++ks) {
        const int k = ks * 4 + hi * 2;
        v2f a = *(const v2f*)(arow + k);         // LDS, conflict-free (stride 66)
        v2f b = *(const v2f*)(wrow + k);
        acc = __builtin_amdgcn_wmma_f32_16x16x4_f32(
            false, a, false, b, (short)0, acc, false, false);
    }

    const float blv = bl[n0 + m];
    const size_t pabs = (size_t)lvl * NODES_PER_LEVEL + p0 + pg * 16;
#pragma unroll
    for (int v = 0; v < 8; ++v) {
        const size_t addr = (pabs + v + 8 * hi) * DIM + n0 + m;
        const float root = out[addr];            // x @ Wr^T term
        out[addr] = tanhf(acc[v] + blv + root);
    }
}

extern "C" void kernel_launch(void* const* d_in, const int* in_sizes, int n_in,
                              void* d_out, int out_size, void* d_ws, size_t ws_size,
                              hipStream_t stream) {
    const float* x    = (const float*)d_in[0];
    const int*   edge = (const int*)d_in[1];     // [2, E]; row 0 = src
    const float* Wl   = (const float*)d_in[2];
    const float* bl   = (const float*)d_in[3];
    const float* Wr   = (const float*)d_in[4];
    float*       out  = (float*)d_out;

    const int N = in_sizes[0] / DIM;             // 131072
    const int* src = edge;                       // first E entries

    // out = x @ Wr^T for every node (root-weight path; also final for level 0)
    dag_root_gemm<<<dim3(N / 128), dim3(256), 0, stream>>>(x, Wr, out);

    // Sequential level sweep; kernel boundaries provide the inter-level sync.
    for (int lvl = 1; lvl < NUM_LEVELS; ++lvl) {
        dag_level<<<dim3(NODES_PER_LEVEL / 32), dim3(256), 0, stream>>>(
            out, src, Wl, bl, lvl);
    }
}